// CosineSimilarity_23673859736169
// MI455X (gfx1250) — compile-verified
//
#include <hip/hip_runtime.h>
#include <hip/hip_bf16.h>
#include <math.h>

#define NQ   30000
#define NSUP 8192
#define CH   128
#define NCLS 16
#define EPSV 1e-8f

typedef __attribute__((ext_vector_type(2))) float v2f;
typedef __attribute__((ext_vector_type(8))) float v8f;

// workspace layout (float offsets)
#define WS_SUMS 0        // 16*128
#define WS_CNT  2048     // 16
#define WS_PN   4096     // 16*128 normalized prototypes
#define WS_PB   8192     // 16*128 protos@Wp + b1
#define WS_QWQ  16384    // 30000*128 Q@Wq

__global__ void k_zero(float* ws) {
    for (int i = threadIdx.x; i < 2064; i += blockDim.x) ws[i] = 0.f;
}

// 64 blocks x 256 threads; each block owns 128 contiguous support rows.
__global__ void __launch_bounds__(256) k_protosum(const float* __restrict__ sf,
                                                  const int* __restrict__ lab,
                                                  float* ws) {
    __shared__ float lsum[NCLS * CH];
    __shared__ float lcnt[NCLS];
    int tid = threadIdx.x;
    for (int i = tid; i < NCLS * CH; i += 256) lsum[i] = 0.f;
    if (tid < NCLS) lcnt[tid] = 0.f;
    __syncthreads();
    int c    = tid & 127;
    int rsel = tid >> 7;
    int start = blockIdx.x * 128;
    for (int r = start + rsel; r < start + 128; r += 2) {
        int l = lab[r];
        atomicAdd(&lsum[l * CH + c], sf[r * CH + c]);
        if (c == 0) atomicAdd(&lcnt[l], 1.f);
    }
    __syncthreads();
    for (int i = tid; i < NCLS * CH; i += 256) atomicAdd(&ws[WS_SUMS + i], lsum[i]);
    if (tid < NCLS) atomicAdd(&ws[WS_CNT + tid], lcnt[tid]);
}

// 1 block x 256: protos, normalized protos (PN), PB = protos@Wp + b1
__global__ void __launch_bounds__(256) k_protofinish(const float* __restrict__ W1,
                                                     const float* __restrict__ b1,
                                                     float* ws) {
    __shared__ float sp[NCLS * CH];
    __shared__ float inv[NCLS];
    int tid = threadIdx.x;
    for (int j = tid; j < NCLS * CH; j += 256) {
        float cnt = ws[WS_CNT + (j >> 7)];
        sp[j] = (cnt > 0.f) ? ws[WS_SUMS + j] / fmaxf(cnt, 1.f) : 0.f;
    }
    __syncthreads();
    if (tid < NCLS) {
        float s = 0.f;
        for (int c = 0; c < CH; ++c) { float v = sp[tid * CH + c]; s += v * v; }
        inv[tid] = 1.f / fmaxf(sqrtf(s), EPSV);
    }
    __syncthreads();
    for (int j = tid; j < NCLS * CH; j += 256) ws[WS_PN + j] = sp[j] * inv[j >> 7];
    for (int j = tid; j < NCLS * CH; j += 256) {
        int n = j >> 7, h = j & 127;
        float acc = b1[h];
        for (int c = 0; c < CH; ++c) acc += sp[n * CH + c] * W1[(CH + c) * CH + h];
        ws[WS_PB + j] = acc;
    }
}

// Per-wave: one 16-query tile. Computes cos (-> cosOut) and Q@Wq (-> ws).
__global__ void __launch_bounds__(256) k_query(const float* __restrict__ qf,
                                               const float* __restrict__ W1,
                                               float* __restrict__ ws,
                                               float* __restrict__ cosOut) {
    extern __shared__ float smem[];
    float* wq  = smem;                 // 128*136 (pad 136: B-reads conflict-free)
    float* pnl = wq + 128 * 136;       // 16*132  (pad 132: A/B-like reads conflict-free)
    int tid = threadIdx.x;
    int wave = tid >> 5, lane = tid & 31;
    float* qst  = pnl + 16 * 132 + wave * (16 * 132 + 16);
    float* invq = qst + 16 * 132;

    for (int i = tid; i < CH * CH; i += 256) {
        int k = i >> 7, h = i & 127;
        wq[k * 136 + h] = W1[i];                 // Wq = W1 rows [0,128)
    }
    for (int i = tid; i < NCLS * CH; i += 256) {
        int n = i >> 7, c = i & 127;
        pnl[n * 132 + c] = ws[WS_PN + i];
    }
    __syncthreads();

    int t = blockIdx.x * 8 + wave;
    if (t >= NQ / 16) return;
    int q0 = t * 16;
    int row = lane & 15, hi = lane >> 4, koff = hi * 2, col = row;

    for (int i = lane; i < 16 * CH; i += 32) {
        int r = i >> 7, c = i & 127;
        qst[r * 132 + c] = qf[(q0 + r) * CH + c];
    }
    if (lane < 16) {
        float s = 0.f;
        for (int c = 0; c < CH; ++c) { float v = qst[lane * 132 + c]; s += v * v; }
        invq[lane] = 1.f / fmaxf(sqrtf(s), EPSV);
    }
    __builtin_amdgcn_wave_barrier();

    // cos tile: A = Q (16xK), B = PN^T (Kx16)
    {
        v8f acc = {};
#pragma unroll
        for (int k0 = 0; k0 < CH; k0 += 4) {
            v2f a = *(const v2f*)&qst[row * 132 + k0 + koff];
            v2f b = *(const v2f*)&pnl[col * 132 + k0 + koff];
            acc = __builtin_amdgcn_wmma_f32_16x16x4_f32(false, a, false, b,
                                                        (short)0, acc, false, false);
        }
#pragma unroll
        for (int r = 0; r < 8; ++r) {
            float val = acc[r] * invq[r + 8 * hi];
            cosOut[(q0 + r + 8 * hi) * NCLS + col] = val;
        }
    }
    // Q@Wq tiles
    for (int n0 = 0; n0 < 8; ++n0) {
        v8f acc = {};
#pragma unroll
        for (int k0 = 0; k0 < CH; k0 += 4) {
            v2f a = *(const v2f*)&qst[row * 132 + k0 + koff];
            v2f b;
            b.x = wq[(k0 + koff) * 136 + n0 * 16 + col];
            b.y = wq[(k0 + koff + 1) * 136 + n0 * 16 + col];
            acc = __builtin_amdgcn_wmma_f32_16x16x4_f32(false, a, false, b,
                                                        (short)0, acc, false, false);
        }
#pragma unroll
        for (int r = 0; r < 8; ++r)
            ws[WS_QWQ + (q0 + r + 8 * hi) * CH + n0 * 16 + col] = acc[r];
    }
}

// Per-wave: one 16-query tile; for each of 16 protos build relu'd h A-frags
// on the fly and run h(16x128) @ W2(128x128) + b2 via WMMA f32 16x16x4.
__global__ void __launch_bounds__(256) k_sim(const float* __restrict__ W2,
                                             const float* __restrict__ W1,
                                             const float* __restrict__ b2,
                                             const float* __restrict__ ws,
                                             const float* __restrict__ cosIn,
                                             float* __restrict__ simOut) {
    extern __shared__ float smem[];
    float* w2l = smem;                 // 128*136
    float* pbl = w2l + 128 * 136;      // 16*132
    float* wcl = pbl + 16 * 132;       // 128
    float* b2l = wcl + 128;            // 128
    int tid = threadIdx.x;
    int wave = tid >> 5, lane = tid & 31;
    float* hst = b2l + 128 + wave * (16 * 132 + 16 * 17);
    float* csl = hst + 16 * 132;       // cos tile [16 rows][stride 17]

    for (int i = tid; i < CH * CH; i += 256) {
        int k = i >> 7, h = i & 127;
        w2l[k * 136 + h] = W2[i];
    }
    for (int i = tid; i < NCLS * CH; i += 256) {
        int n = i >> 7, c = i & 127;
        pbl[n * 132 + c] = ws[WS_PB + i];
    }
    if (tid < CH) {
        wcl[tid] = W1[2 * CH * CH + tid];   // Wc = W1 row 256
        b2l[tid] = b2[tid];
    }
    __syncthreads();

    int t = blockIdx.x * 8 + wave;
    if (t >= NQ / 16) return;
    int q0 = t * 16;
    int row = lane & 15, hi = lane >> 4, koff = hi * 2, col = row;

    for (int i = lane; i < 16 * CH; i += 32) {
        int r = i >> 7, c = i & 127;
        hst[r * 132 + c] = ws[WS_QWQ + (q0 + r) * CH + c];
    }
    for (int i = lane; i < 256; i += 32) {
        int r = i >> 4, n = i & 15;
        csl[r * 17 + n] = cosIn[(q0 + r) * NCLS + n];
    }
    __builtin_amdgcn_wave_barrier();

    float b2v[8];
#pragma unroll
    for (int n0 = 0; n0 < 8; ++n0) b2v[n0] = b2l[n0 * 16 + col];

    for (int n = 0; n < NCLS; ++n) {
        float cosv = csl[row * 17 + n];
        v8f acc[8];
#pragma unroll
        for (int n0 = 0; n0 < 8; ++n0) {
#pragma unroll
            for (int r = 0; r < 8; ++r) acc[n0][r] = b2v[n0];
        }
#pragma unroll 4
        for (int k0 = 0; k0 < CH; k0 += 4) {
            v2f qw = *(const v2f*)&hst[row * 132 + k0 + koff];
            v2f pb = *(const v2f*)&pbl[n * 132 + k0 + koff];
            v2f wc = *(const v2f*)&wcl[k0 + koff];
            v2f a;
            a.x = fmaxf(fmaf(cosv, wc.x, qw.x + pb.x), 0.f);
            a.y = fmaxf(fmaf(cosv, wc.y, qw.y + pb.y), 0.f);
#pragma unroll
            for (int n0 = 0; n0 < 8; ++n0) {
                v2f b;
                b.x = w2l[(k0 + koff) * 136 + n0 * 16 + col];
                b.y = w2l[(k0 + koff + 1) * 136 + n0 * 16 + col];
                acc[n0] = __builtin_amdgcn_wmma_f32_16x16x4_f32(false, a, false, b,
                                                                (short)0, acc[n0],
                                                                false, false);
            }
        }
#pragma unroll
        for (int n0 = 0; n0 < 8; ++n0) {
#pragma unroll
            for (int r = 0; r < 8; ++r) {
                simOut[((q0 + r + 8 * hi) * NCLS + n) * CH + n0 * 16 + col] = acc[n0][r];
            }
        }
    }
}

extern "C" void kernel_launch(void* const* d_in, const int* in_sizes, int n_in,
                              void* d_out, int out_size, void* d_ws, size_t ws_size,
                              hipStream_t stream) {
    const float* sf  = (const float*)d_in[0];
    const int*   lab = (const int*)d_in[1];
    const float* qf  = (const float*)d_in[2];
    const float* W1  = (const float*)d_in[3];
    const float* b1  = (const float*)d_in[4];
    const float* W2  = (const float*)d_in[5];
    const float* b2  = (const float*)d_in[6];
    float* out    = (float*)d_out;
    float* cosOut = out + (size_t)NQ * NCLS * CH;   // cos region of d_out
    float* ws     = (float*)d_ws;

    k_zero<<<1, 256, 0, stream>>>(ws);
    k_protosum<<<NSUP / 128, 256, 0, stream>>>(sf, lab, ws);
    k_protofinish<<<1, 256, 0, stream>>>(W1, b1, ws);

    int nt = NQ / 16;                 // 1875 query tiles
    int blocks = (nt + 7) / 8;        // 8 waves/block, 1 tile/wave
    size_t smemQ = (size_t)(128 * 136 + 16 * 132 + 8 * (16 * 132 + 16)) * 4;
    size_t smemC = (size_t)(128 * 136 + 16 * 132 + 128 + 128 +
                            8 * (16 * 132 + 16 * 17)) * 4;
    k_query<<<blocks, 256, smemQ, stream>>>(qf, W1, ws, cosOut);
    k_sim<<<blocks, 256, smemC, stream>>>(W2, W1, b2, ws, cosOut, out);
}